// TopologyGCNLayer_75995151335922
// MI455X (gfx1250) — compile-verified
//
#include <hip/hip_runtime.h>

#define N_DIM 128
#define LN_EPS 1e-5f

typedef __attribute__((ext_vector_type(2))) float v2f;
typedef __attribute__((ext_vector_type(8))) float v8f;

// ---------------------------------------------------------------------------
// Zero workspace (deg + neigh), float4-vectorized
// ---------------------------------------------------------------------------
__global__ void zero_kernel(float4* __restrict__ p, long long n4) {
    long long i = (long long)blockIdx.x * blockDim.x + threadIdx.x;
    if (i < n4) p[i] = make_float4(0.f, 0.f, 0.f, 0.f);
}

// ---------------------------------------------------------------------------
// deg[src[e]] += 1  (exact in f32: E = 1.6M < 2^24)
// ---------------------------------------------------------------------------
__global__ void deg_kernel(const int* __restrict__ src, float* __restrict__ deg,
                           int n_edges) {
    int e = blockIdx.x * blockDim.x + threadIdx.x;
    if (e < n_edges) unsafeAtomicAdd(&deg[src[e]], 1.0f);
}

// deg[i] -> 1 / max(deg[i], 1)
__global__ void invdeg_kernel(float* __restrict__ deg, int n) {
    int i = blockIdx.x * blockDim.x + threadIdx.x;
    if (i < n) deg[i] = 1.0f / fmaxf(deg[i], 1.0f);
}

// ---------------------------------------------------------------------------
// Edge scatter: neigh[src] += x[dst] * inv_deg[dst].
// One wave per edge; lane handles 4 consecutive floats (32*4 = 128 cols).
// x and neigh are L2-resident (51.2 MB each vs 192 MB L2).
// ---------------------------------------------------------------------------
__global__ __launch_bounds__(256) void scatter_kernel(
    const float* __restrict__ x, const int* __restrict__ src,
    const int* __restrict__ dst, const float* __restrict__ inv_deg,
    float* __restrict__ neigh, int n_edges) {
    int e = blockIdx.x * 8 + (threadIdx.x >> 5);
    if (e >= n_edges) return;
    int lane = threadIdx.x & 31;
    int s = src[e];
    int d = dst[e];
    float w = inv_deg[d];
    const float4 xv = *(const float4*)(x + (size_t)d * N_DIM + lane * 4);
    float* np = neigh + (size_t)s * N_DIM + lane * 4;
    unsafeAtomicAdd(np + 0, xv.x * w);
    unsafeAtomicAdd(np + 1, xv.y * w);
    unsafeAtomicAdd(np + 2, xv.z * w);
    unsafeAtomicAdd(np + 3, xv.w * w);
}

// ---------------------------------------------------------------------------
// Fused:  h = x + neigh @ W^T + b ;  out = LayerNorm(h) * gamma + beta
// Block = 256 threads = 8 waves; block owns 16 rows (nodes).
// Wave w computes the 16x16 output tile at cols [16w, 16w+16) with
// V_WMMA_F32_16X16X4_F32 over K=128 (32 steps), full f32 precision.
// Epilogue: tile (+x +b) -> 16x128 floats in LDS; then row-wise LN with 16 threads
// per row (shfl_xor width-16 reduction, aligned to wave32 halves).
// ---------------------------------------------------------------------------
__global__ __launch_bounds__(256) void gemm_ln_kernel(
    const float* __restrict__ x, const float* __restrict__ W,
    const float* __restrict__ bias, const float* __restrict__ gamma,
    const float* __restrict__ beta, const float* __restrict__ neigh,
    float* __restrict__ out) {
    __shared__ float hsm[16 * N_DIM];

    const int tid = threadIdx.x;
    const int wave = tid >> 5;   // 0..7 -> column tile
    const int lane = tid & 31;
    const int half = lane >> 4;  // 0 or 1
    const int l15 = lane & 15;
    const int row_base = blockIdx.x * 16;
    const int col_base = wave * 16;

    // A (16x4 f32): lane l -> m = l&15, k = 2*(l>>4) + {0,1}  (float2 load)
    // B (4x16) = W^T tile: B[k][n] = W[col_base+n][k]; symmetric layout.
    const float* aptr = neigh + (size_t)(row_base + l15) * N_DIM + half * 2;
    const float* bptr = W + (size_t)(col_base + l15) * N_DIM + half * 2;

    v8f c = {};
#pragma unroll
    for (int k0 = 0; k0 < N_DIM; k0 += 4) {
        v2f a = *(const v2f*)(aptr + k0);
        v2f b = *(const v2f*)(bptr + k0);
        c = __builtin_amdgcn_wmma_f32_16x16x4_f32(
            /*neg_a=*/false, a, /*neg_b=*/false, b,
            /*c_mod=*/(short)0, c, /*reuse_a=*/false, /*reuse_b=*/false);
    }

    // D layout: VGPR r -> m = r + 8*half, n = l15. Add residual + bias -> LDS.
#pragma unroll
    for (int r = 0; r < 8; ++r) {
        int m = r + half * 8;
        int col = col_base + l15;
        float h = c[r] + x[(size_t)(row_base + m) * N_DIM + col] + bias[col];
        hsm[m * N_DIM + col] = h;
    }
    __syncthreads();

    // LayerNorm: row = tid>>4 (lanes 0-15 / 16-31 of each wave = one row each)
    const int row = tid >> 4;
    const int cg = (tid & 15) * 8;  // 8 columns per thread
    float s = 0.f, q = 0.f;
#pragma unroll
    for (int j = 0; j < 8; ++j) {
        float v = hsm[row * N_DIM + cg + j];
        s += v;
        q += v * v;
    }
#pragma unroll
    for (int off = 8; off >= 1; off >>= 1) {
        s += __shfl_xor(s, off, 16);
        q += __shfl_xor(q, off, 16);
    }
    const float mean = s * (1.0f / N_DIM);
    const float var = q * (1.0f / N_DIM) - mean * mean;
    const float inv = rsqrtf(var + LN_EPS);
    const size_t obase = (size_t)(row_base + row) * N_DIM;
#pragma unroll
    for (int j = 0; j < 8; ++j) {
        int cidx = cg + j;
        float v = hsm[row * N_DIM + cidx];
        out[obase + cidx] = (v - mean) * inv * gamma[cidx] + beta[cidx];
    }
}

// ---------------------------------------------------------------------------
extern "C" void kernel_launch(void* const* d_in, const int* in_sizes, int n_in,
                              void* d_out, int out_size, void* d_ws,
                              size_t ws_size, hipStream_t stream) {
    const float* x     = (const float*)d_in[0];  // [1, N, 128]
    const float* W     = (const float*)d_in[1];  // [128, 128]
    const float* bias  = (const float*)d_in[2];  // [128]
    const float* gamma = (const float*)d_in[3];  // [128]
    const float* beta  = (const float*)d_in[4];  // [128]
    const int* edge    = (const int*)d_in[5];    // [2, E]

    const int n_edges = in_sizes[5] / 2;
    const int n_nodes = in_sizes[0] / N_DIM;  // B = 1
    const int* src = edge;
    const int* dst = edge + n_edges;

    float* deg   = (float*)d_ws;          // N floats (becomes inv_deg)
    float* neigh = deg + n_nodes;         // N * 128 floats
    float* out   = (float*)d_out;

    // 1) zero deg + neigh (contiguous region, divisible by 4 floats)
    long long n4 = ((long long)n_nodes * (N_DIM + 1)) / 4;
    zero_kernel<<<(int)((n4 + 255) / 256), 256, 0, stream>>>((float4*)d_ws, n4);

    // 2) out-degree as src
    deg_kernel<<<(n_edges + 255) / 256, 256, 0, stream>>>(src, deg, n_edges);

    // 3) inv_deg = 1 / max(deg, 1)
    invdeg_kernel<<<(n_nodes + 255) / 256, 256, 0, stream>>>(deg, n_nodes);

    // 4) scatter: one wave per edge, 8 edges per block
    scatter_kernel<<<(n_edges + 7) / 8, 256, 0, stream>>>(x, src, dst, deg,
                                                          neigh, n_edges);

    // 5) fused WMMA projection + residual + bias + LayerNorm
    //    N = 100000 is divisible by 16 -> 6250 full blocks, EXEC all-ones.
    gemm_ln_kernel<<<n_nodes / 16, 256, 0, stream>>>(x, W, bias, gamma, beta,
                                                     neigh, out);
}